// ROIPoolingLayer_76888504533182
// MI455X (gfx1250) — compile-verified
//
#include <hip/hip_runtime.h>
#include <cstddef>

// ROI crop-and-resize bilinear pooling for MI455X (gfx1250).
// Memory-bound gather (AI ~1.4 flop/byte -> no WMMA; HBM floor ~7us from the
// 103MB streaming write; gathers served by the 192MB L2 holding the 64MB fm).
// gfx1250 paths: GLOBAL_LOAD_ASYNC_TO_LDS_B128 triple-buffered software
// pipeline (ASYNCcnt / s_wait_asynccnt), NT streaming stores, global_prefetch
// in the fallback path.

typedef float f4  __attribute__((ext_vector_type(4)));
typedef int   v4i __attribute__((vector_size(16)));   // matches builtin param

#define POOL   7
#define BATCH  8
#define FH     64
#define FW     64
#define FC     512
#define NR     128
#define C4     (FC / 4)        // 128 float4 per pixel
#define CELLS  (POOL * POOL)   // 49
#define UNITS  (CELLS * C4)    // 6272 float4 outputs per ROI
#define NTHR   256

#if defined(__HIP_DEVICE_COMPILE__) && \
    __has_builtin(__builtin_amdgcn_global_load_async_to_lds_b128) && \
    __has_builtin(__builtin_amdgcn_s_wait_asynccnt)
#define USE_ASYNC 1
// addrspace-qualified vector pointer types required by the builtin
// (clang prints AS1 as "__device__", AS3 as "__shared__")
typedef __attribute__((address_space(1))) v4i* as1_v4i_p;
typedef __attribute__((address_space(3))) v4i* as3_v4i_p;
#else
#define USE_ASYNC 0
#endif

__device__ __forceinline__ int iclamp(int v, int lo, int hi) {
    return v < lo ? lo : (v > hi ? hi : v);
}

__global__ __launch_bounds__(NTHR)
void ROIPoolingLayer_76888504533182_kernel(const float* __restrict__ fm,
                                           const float* __restrict__ rois,
                                           float* __restrict__ out)
{
    __shared__ int4   s_off[CELLS];   // float4-unit offsets of the 4 taps
    __shared__ float2 s_w[CELLS];     // (wy, wx)
    __shared__ int    s_a0[2 * POOL];
    __shared__ int    s_a1[2 * POOL];
    __shared__ float  s_fw[2 * POOL];

    const int t   = threadIdx.x;
    const int roi = blockIdx.x;        // b * NR + r, 0..1023
    const int b   = roi >> 7;          // NR == 128

    // ---- phase 1: axis coordinates (t=0..6 -> y, t=7..13 -> x) ----
    if (t < 2 * POOL) {
        const int axis = t / POOL;     // 0 = y (rows), 1 = x (cols)
        const int i    = t % POOL;
        const float* rp = rois + (size_t)roi * 4;
        const float lo_n = rp[axis];        // y1 or x1 (normalized)
        const float hi_n = rp[axis + 2];    // y2 or x2
        const int   size = 64;              // FH == FW == 64
        int base = iclamp((int)(lo_n * 64.0f), 0, size);   // truncating cast
        int e    = iclamp((int)(hi_n * 64.0f), 0, size);
        int c    = e - base; if (c < 1) c = 1;
        float s   = ((float)i + 0.5f) * ((float)c / 7.0f) - 0.5f;
        float cm1 = (float)c - 1.0f;
        s = fminf(fmaxf(s, 0.0f), cm1);
        int   i0 = (int)floorf(s);
        float w  = s - (float)i0;
        int a0 = iclamp(base + iclamp(i0,     0, c - 1), 0, size - 1);
        int a1 = iclamp(base + iclamp(i0 + 1, 0, c - 1), 0, size - 1);
        s_a0[t] = a0; s_a1[t] = a1; s_fw[t] = w;
    }
    __syncthreads();

    // ---- phase 2: pack per-cell tap offsets + weights ----
    if (t < CELLS) {
        const int i = t / POOL, j = t % POOL;
        const int ya0 = s_a0[i],        ya1 = s_a1[i];
        const int xa0 = s_a0[POOL + j], xa1 = s_a1[POOL + j];
        int4 o;
        o.x = (ya0 * FW + xa0) * C4;   // v00
        o.y = (ya0 * FW + xa1) * C4;   // v01
        o.z = (ya1 * FW + xa0) * C4;   // v10
        o.w = (ya1 * FW + xa1) * C4;   // v11
        s_off[t] = o;
        s_w[t]   = make_float2(s_fw[i], s_fw[POOL + j]);   // (wy, wx)
    }
    __syncthreads();

    const f4* __restrict__ fmb  = (const f4*)fm + (size_t)b * (FH * FW * C4);
    f4*       __restrict__ outb = (f4*)out + (size_t)roi * UNITS;

#if USE_ASYNC
    // ---- triple-buffered async pipeline: global -> LDS 2 iterations ahead.
    // Each thread consumes only its own staged slots, so the only sync needed
    // is ASYNCcnt (async loads complete in order).
    __shared__ f4 s_stage[3][4][NTHR];   // 48 KB of the 320 KB WGP LDS

    auto issue = [&](int u, int stg) {
        const int  c4 = u & (C4 - 1);
        const int4 o  = s_off[u >> 7];
        __builtin_amdgcn_global_load_async_to_lds_b128(
            (as1_v4i_p)(fmb + o.x + c4), (as3_v4i_p)&s_stage[stg][0][t], 0, 0);
        __builtin_amdgcn_global_load_async_to_lds_b128(
            (as1_v4i_p)(fmb + o.y + c4), (as3_v4i_p)&s_stage[stg][1][t], 0, 0);
        __builtin_amdgcn_global_load_async_to_lds_b128(
            (as1_v4i_p)(fmb + o.z + c4), (as3_v4i_p)&s_stage[stg][2][t], 0, 0);
        __builtin_amdgcn_global_load_async_to_lds_b128(
            (as1_v4i_p)(fmb + o.w + c4), (as3_v4i_p)&s_stage[stg][3][t], 0, 0);
    };

    issue(t, 0);               // iteration 0
    issue(t + NTHR, 1);        // iteration 1 (t+256 < 6272 always)
    int buf = 0;

    for (int u = t; u < UNITS; u += NTHR) {
        const int u2 = u + 2 * NTHR;
        if (u2 < UNITS) {
            int nb = buf + 2; if (nb >= 3) nb -= 3;
            issue(u2, nb);
            __builtin_amdgcn_s_wait_asynccnt(8);   // oldest 4 (this iter) done
        } else if (u + NTHR < UNITS) {
            __builtin_amdgcn_s_wait_asynccnt(4);
        } else {
            __builtin_amdgcn_s_wait_asynccnt(0);
        }

        const float2 wv = s_w[u >> 7];
        const f4 v00 = s_stage[buf][0][t];
        const f4 v01 = s_stage[buf][1][t];
        const f4 v10 = s_stage[buf][2][t];
        const f4 v11 = s_stage[buf][3][t];

        const float wy = wv.x, wx = wv.y;
        const float omwx = 1.0f - wx, omwy = 1.0f - wy;
        const f4 top = v00 * omwx + v01 * wx;
        const f4 bot = v10 * omwx + v11 * wx;
        const f4 res = top * omwy + bot * wy;
        __builtin_nontemporal_store(res, outb + u);

        buf += 1; if (buf >= 3) buf = 0;
    }
#else
    // ---- fallback: direct b128 gathers + gfx1250 global_prefetch ----
    #pragma unroll 2
    for (int u = t; u < UNITS; u += NTHR) {
        const int cell = u >> 7;
        const int c4   = u & (C4 - 1);
        const int4   o  = s_off[cell];
        const float2 wv = s_w[cell];

        const int un = u + NTHR;
        if (un < UNITS) {
            const int4 on  = s_off[un >> 7];
            const int  c4n = un & (C4 - 1);
            __builtin_prefetch((const void*)(fmb + on.x + c4n), 0, 3);
            __builtin_prefetch((const void*)(fmb + on.y + c4n), 0, 3);
            __builtin_prefetch((const void*)(fmb + on.z + c4n), 0, 3);
            __builtin_prefetch((const void*)(fmb + on.w + c4n), 0, 3);
        }

        const f4 v00 = fmb[o.x + c4];
        const f4 v01 = fmb[o.y + c4];
        const f4 v10 = fmb[o.z + c4];
        const f4 v11 = fmb[o.w + c4];

        const float wy = wv.x, wx = wv.y;
        const float omwx = 1.0f - wx, omwy = 1.0f - wy;
        const f4 top = v00 * omwx + v01 * wx;
        const f4 bot = v10 * omwx + v11 * wx;
        const f4 res = top * omwy + bot * wy;
        __builtin_nontemporal_store(res, outb + u);
    }
#endif
}

extern "C" void kernel_launch(void* const* d_in, const int* in_sizes, int n_in,
                              void* d_out, int out_size, void* d_ws, size_t ws_size,
                              hipStream_t stream) {
    const float* fm   = (const float*)d_in[0];   // [8,64,64,512] f32
    const float* rois = (const float*)d_in[1];   // [8,128,4]     f32
    float*       out  = (float*)d_out;           // [8,128,7,7,512] f32
    (void)in_sizes; (void)n_in; (void)out_size; (void)d_ws; (void)ws_size;

    dim3 grid(BATCH * NR);   // one 8-wave32 block per ROI
    dim3 block(NTHR);
    ROIPoolingLayer_76888504533182_kernel<<<grid, block, 0, stream>>>(fm, rois, out);
}